// Net_67680094650527
// MI455X (gfx1250) — compile-verified
//
#include <hip/hip_runtime.h>
#include <hip/hip_bf16.h>

typedef __attribute__((ext_vector_type(2))) float v2f;
typedef __attribute__((ext_vector_type(8))) float v8f;

// ---------------- utility kernels ----------------

__global__ void zero_f32_kernel(float* __restrict__ p, long n) {
    long i = (long)blockIdx.x * blockDim.x + threadIdx.x;
    long stride = (long)gridDim.x * blockDim.x;
    for (; i < n; i += stride) p[i] = 0.0f;
}

__global__ void deg_kernel(const long long* __restrict__ src,
                           const float* __restrict__ w,
                           float* __restrict__ deg, int E) {
    int e = blockIdx.x * blockDim.x + threadIdx.x;
    if (e >= E) return;
    unsafeAtomicAdd(&deg[(int)src[e]], w[e]);
}

__global__ void dinv_kernel(float* __restrict__ deg, int n) {
    int i = blockIdx.x * blockDim.x + threadIdx.x;
    if (i >= n) return;
    float d = deg[i];
    deg[i] = (d > 0.0f) ? rsqrtf(fmaxf(d, 1e-30f)) : 0.0f;
}

__global__ void norm_kernel(const long long* __restrict__ src,
                            const long long* __restrict__ dst,
                            const float* __restrict__ w,
                            const float* __restrict__ dinv,
                            float* __restrict__ norm, int E) {
    int e = blockIdx.x * blockDim.x + threadIdx.x;
    if (e >= E) return;
    norm[e] = -dinv[(int)src[e]] * w[e] * dinv[(int)dst[e]];
}

// dst[i*KP + k] = (k < F) ? src[i*F + k] : 0
template <int F, int KP>
__global__ void pad_copy_kernel(const float* __restrict__ src,
                                float* __restrict__ dst, int n) {
    long i = (long)blockIdx.x * blockDim.x + threadIdx.x;
    long total = (long)n * KP;
    if (i >= total) return;
    int k = (int)(i % KP);
    long row = i / KP;
    dst[i] = (k < F) ? src[row * F + k] : 0.0f;
}

// Wp[(t*NP + nn)*KP + k] = W[(t*FIN + k)*FOUT + nn]  (zero-padded, transposed)
template <int FIN, int FOUT, int KP, int NP>
__global__ void prep_w_kernel(const float* __restrict__ W,
                              float* __restrict__ Wp) {
    int i = blockIdx.x * blockDim.x + threadIdx.x;
    const int total = 3 * NP * KP;
    if (i >= total) return;
    int k = i % KP;
    int nn = (i / KP) % NP;
    int t = i / (KP * NP);
    Wp[i] = (k < FIN && nn < FOUT) ? W[((size_t)t * FIN + k) * FOUT + nn] : 0.0f;
}

// out[dst*FP + f] += norm * h[src*FP + f]  for f < F   (out pre-zeroed, stride FP)
template <int F, int FP>
__global__ void prop_kernel(const long long* __restrict__ src,
                            const long long* __restrict__ dst,
                            const float* __restrict__ norm,
                            const float* __restrict__ h,
                            float* __restrict__ out, int E) {
    int e = blockIdx.x * blockDim.x + threadIdx.x;
    if (e >= E) return;
    float nv = norm[e];
    size_t s = (size_t)src[e] * FP;
    size_t d = (size_t)dst[e] * FP;
#pragma unroll
    for (int f = 0; f < F; ++f) {
        unsafeAtomicAdd(&out[d + f], nv * h[s + f]);
    }
}

// T2 = 2*P2 - X  (elementwise over n*KP; pads stay zero)
__global__ void tx2_kernel(const float* __restrict__ P2,
                           const float* __restrict__ X,
                           float* __restrict__ T2, long total) {
    long i = (long)blockIdx.x * blockDim.x + threadIdx.x;
    if (i >= total) return;
    T2[i] = 2.0f * P2[i] - X[i];
}

// ---------------- WMMA Chebyshev combine ----------------
// out = X@W0 + T1@W1 + T2@W2 + bias (optional ReLU), all operands pre-padded:
//   X/T1/T2: n rows, stride KP (KP % 4 == 0, zero-padded)
//   Wp: [t][NP cols][KP] transposed, zero-padded
// One wave32 per 16x16 C tile, V_WMMA_F32_16X16X4_F32, K stepped by 4.
// ISA 7.12.2 layouts: A/B vgpr v, half-wave g: K = k0 + v + 2g (contiguous pair
// per lane -> one b64 load); C vgpr v: M = v + 8g, N = lane%16.
template <int KP, int FOUT, int NT, int OSTRIDE, bool RELU>
__global__ void cheb_combine_kernel(const float* __restrict__ Xp,
                                    const float* __restrict__ T1,
                                    const float* __restrict__ T2,
                                    const float* __restrict__ Wp,
                                    const float* __restrict__ bias,
                                    float* __restrict__ out, int n) {
    constexpr int NP = NT * 16;
    const int lane = threadIdx.x & 31;
    const int gwave = blockIdx.x * (blockDim.x >> 5) + (threadIdx.x >> 5);
    const int mTiles = (n + 15) / 16;
    if (gwave >= mTiles * NT) return;            // uniform per wave
    const int mt = gwave / NT;
    const int nt = gwave - mt * NT;
    const int row0 = mt * 16;
    const int n0 = nt * 16;
    const int g = lane >> 4;                     // half-wave group
    const int lm = lane & 15;

    const int arow = min(row0 + lm, n - 1);      // clamp (no branch) for M tail
    const int nn = n0 + lm;                      // < NP always
    const size_t aoff = (size_t)arow * KP + 2 * g;

    const float* a0 = Xp + aoff;
    const float* a1 = T1 + aoff;
    const float* a2 = T2 + aoff;
    const float* bp = Wp + (size_t)nn * KP + 2 * g;   // term 0; +NP*KP per term

    v8f acc = {0.f, 0.f, 0.f, 0.f, 0.f, 0.f, 0.f, 0.f};

#pragma unroll
    for (int t = 0; t < 3; ++t) {
        const float* Ap = (t == 0) ? a0 : ((t == 1) ? a1 : a2);
        const float* Bp = bp + (size_t)t * NP * KP;
#pragma unroll
        for (int k0 = 0; k0 < KP; k0 += 4) {
            v2f a = *(const v2f*)(Ap + k0);
            v2f b = *(const v2f*)(Bp + k0);
            acc = __builtin_amdgcn_wmma_f32_16x16x4_f32(
                false, a, false, b, (short)0, acc, false, false);
        }
    }

    // epilogue (only divergence in the kernel)
    if (nn < OSTRIDE) {
        float bv = (nn < FOUT) ? bias[nn] : 0.0f;
#pragma unroll
        for (int v = 0; v < 8; ++v) {
            int row = row0 + v + 8 * g;
            if (row < n) {
                float val = acc[v] + bv;
                if (RELU) val = fmaxf(val, 0.0f);
                if (nn >= FOUT) val = 0.0f;      // zero the pad columns
                out[(size_t)row * OSTRIDE + nn] = val;
            }
        }
    }
}

// in-place log_softmax over 4 columns
__global__ void log_softmax4_kernel(float* __restrict__ io, int n) {
    int i = blockIdx.x * blockDim.x + threadIdx.x;
    if (i >= n) return;
    float v0 = io[i * 4 + 0], v1 = io[i * 4 + 1];
    float v2 = io[i * 4 + 2], v3 = io[i * 4 + 3];
    float m = fmaxf(fmaxf(v0, v1), fmaxf(v2, v3));
    float s = __expf(v0 - m) + __expf(v1 - m) + __expf(v2 - m) + __expf(v3 - m);
    float l = m + __logf(s);
    io[i * 4 + 0] = v0 - l;
    io[i * 4 + 1] = v1 - l;
    io[i * 4 + 2] = v2 - l;
    io[i * 4 + 3] = v3 - l;
}

// ---------------- host launcher ----------------

static inline int cdiv(long a, int b) { return (int)((a + b - 1) / b); }

extern "C" void kernel_launch(void* const* d_in, const int* in_sizes, int n_in,
                              void* d_out, int out_size, void* d_ws, size_t ws_size,
                              hipStream_t stream) {
    const float*     x   = (const float*)d_in[0];
    const long long* ei  = (const long long*)d_in[1];   // int64 (2, E)
    const float*     ew  = (const float*)d_in[2];
    const float*     W1  = (const float*)d_in[3];
    const float*     b1  = (const float*)d_in[4];
    const float*     W2  = (const float*)d_in[5];
    const float*     b2  = (const float*)d_in[6];
    const float*     W3  = (const float*)d_in[7];
    const float*     b3  = (const float*)d_in[8];

    const int n = in_sizes[0] / 33;        // 100000
    const int E = in_sizes[2];             // 3200000
    const long long* src = ei;
    const long long* dst = ei + E;

    // padded strides: KP1=36 (33), KP2=28 (25), KP3=16 (16)
    constexpr int KP1 = 36, KP2 = 28, KP3 = 16;

    // workspace layout (floats)
    float* ws   = (float*)d_ws;
    float* dinv = ws;                               // n
    float* norm = dinv + n;                         // E
    float* xp   = norm + E;                         // n*36 (padded x)
    float* bufA = xp   + (size_t)n * KP1;           // n*36 (T1)
    float* bufB = bufA + (size_t)n * KP1;           // n*36 (P2)
    float* bufC = bufB + (size_t)n * KP1;           // n*36 (T2)
    float* h1   = bufC + (size_t)n * KP1;           // n*28 (stride KP2)
    float* h2   = h1   + (size_t)n * KP2;           // n*16 (stride KP3)
    float* Wp1  = h2   + (size_t)n * KP3;           // 3*32*36
    float* Wp2  = Wp1  + 3 * 32 * KP1;              // 3*16*28
    float* Wp3  = Wp2  + 3 * 16 * KP2;              // 3*16*16
    float* outF = (float*)d_out;                    // n*4

    const int B = 256;
    const int gE = cdiv(E, B);
    const int gN = cdiv(n, B);
    const int wavesPerBlock = B / 32;
    const int mTiles = (n + 15) / 16;

    // ---- normalization ----
    zero_f32_kernel<<<gN, B, 0, stream>>>(dinv, (long)n);
    deg_kernel<<<gE, B, 0, stream>>>(src, ew, dinv, E);
    dinv_kernel<<<gN, B, 0, stream>>>(dinv, n);
    norm_kernel<<<gE, B, 0, stream>>>(src, dst, ew, dinv, norm, E);

    // ---- prep: padded x, padded/transposed weights ----
    pad_copy_kernel<33, KP1><<<cdiv((long)n * KP1, B), B, 0, stream>>>(x, xp, n);
    prep_w_kernel<33, 25, KP1, 32><<<cdiv(3 * 32 * KP1, B), B, 0, stream>>>(W1, Wp1);
    prep_w_kernel<25, 16, KP2, 16><<<cdiv(3 * 16 * KP2, B), B, 0, stream>>>(W2, Wp2);
    prep_w_kernel<16,  4, KP3, 16><<<cdiv(3 * 16 * KP3, B), B, 0, stream>>>(W3, Wp3);

    // ---- layer 1: 33 -> 25, ReLU ----
    zero_f32_kernel<<<cdiv((long)n * KP1, B), B, 0, stream>>>(bufA, (long)n * KP1);
    prop_kernel<33, KP1><<<gE, B, 0, stream>>>(src, dst, norm, xp, bufA, E);
    zero_f32_kernel<<<cdiv((long)n * KP1, B), B, 0, stream>>>(bufB, (long)n * KP1);
    prop_kernel<33, KP1><<<gE, B, 0, stream>>>(src, dst, norm, bufA, bufB, E);
    tx2_kernel<<<cdiv((long)n * KP1, B), B, 0, stream>>>(bufB, xp, bufC, (long)n * KP1);
    cheb_combine_kernel<KP1, 25, 2, KP2, true>
        <<<cdiv(mTiles * 2, wavesPerBlock), B, 0, stream>>>(xp, bufA, bufC, Wp1, b1, h1, n);

    // ---- layer 2: 25 -> 16 ----
    zero_f32_kernel<<<cdiv((long)n * KP2, B), B, 0, stream>>>(bufA, (long)n * KP2);
    prop_kernel<25, KP2><<<gE, B, 0, stream>>>(src, dst, norm, h1, bufA, E);
    zero_f32_kernel<<<cdiv((long)n * KP2, B), B, 0, stream>>>(bufB, (long)n * KP2);
    prop_kernel<25, KP2><<<gE, B, 0, stream>>>(src, dst, norm, bufA, bufB, E);
    tx2_kernel<<<cdiv((long)n * KP2, B), B, 0, stream>>>(bufB, h1, bufC, (long)n * KP2);
    cheb_combine_kernel<KP2, 16, 1, KP3, false>
        <<<cdiv(mTiles, wavesPerBlock), B, 0, stream>>>(h1, bufA, bufC, Wp2, b2, h2, n);

    // ---- layer 3: 16 -> 4 ----
    zero_f32_kernel<<<cdiv((long)n * KP3, B), B, 0, stream>>>(bufA, (long)n * KP3);
    prop_kernel<16, KP3><<<gE, B, 0, stream>>>(src, dst, norm, h2, bufA, E);
    zero_f32_kernel<<<cdiv((long)n * KP3, B), B, 0, stream>>>(bufB, (long)n * KP3);
    prop_kernel<16, KP3><<<gE, B, 0, stream>>>(src, dst, norm, bufA, bufB, E);
    tx2_kernel<<<cdiv((long)n * KP3, B), B, 0, stream>>>(bufB, h2, bufC, (long)n * KP3);
    cheb_combine_kernel<KP3, 4, 1, 4, false>
        <<<cdiv(mTiles, wavesPerBlock), B, 0, stream>>>(h2, bufA, bufC, Wp3, b3, outF, n);

    // ---- log_softmax over 4 classes ----
    log_softmax4_kernel<<<gN, B, 0, stream>>>(outF, n);
}